// G2CoreAttention_6674379178219
// MI455X (gfx1250) — compile-verified
//
#include <hip/hip_runtime.h>

typedef __attribute__((ext_vector_type(2))) float    v2f;
typedef __attribute__((ext_vector_type(8))) float    v8f;
typedef __attribute__((ext_vector_type(4))) int      v4i;
typedef __attribute__((ext_vector_type(8))) int      v8i;
typedef __attribute__((ext_vector_type(4))) unsigned v4u;

namespace {
constexpr int kS       = 2048;
constexpr int kH       = 16;
constexpr int kD       = 128;
constexpr int kNKV     = 8192;
constexpr int kTopK    = 512;
constexpr int kKTile   = 16;
constexpr int kTiles   = kTopK / kKTile;    // 32 key tiles
constexpr int kDChunk  = kD / 4;            // 32 K=4 chunks along D
constexpr int kDTiles  = kD / 16;           // 8 output d-tiles
constexpr int kWaves   = 4;
constexpr int kPitch   = 140;               // LDS row pitch in dwords (128 + 12 pad)
constexpr int kBufF    = kKTile * kPitch;   // 2240 floats per tile buffer
constexpr int kSmemF   = kWaves * 2 * kBufF;// 17920 floats = 70 KB (merge aliases here)
}

// Issue one TDM gather: 16 indexed rows of kv[8192][128] f32 -> LDS (pitch 140 dw).
// D# packing per CDNA5 ISA ch.8 (gather mode, 16-bit indices, pad 12 dw / 128 dw).
// clang-23 toolchain: 6-arg builtin (extra int32x8 before cpol; zero-filled).
__device__ __forceinline__ void tdm_gather16(unsigned ldsByteOff,
                                             const float* kvBase,
                                             const int* rows) {
  unsigned long long ga = (unsigned long long)(uintptr_t)kvBase;
  v4u g0;
  g0.x = 0x80000001u;                                   // gather_mode=1, idx16, count=1
  g0.y = (unsigned)__builtin_amdgcn_readfirstlane((int)ldsByteOff);
  g0.z = (unsigned)(ga & 0xffffffffull);
  g0.w = (unsigned)((ga >> 32) & 0x1ffffffull) | 0x80000000u;  // type=2 ("image")
  v8i g1;
  g1[0] = (int)0x17920000;        // pad_amount=12dw(11), pad_interval=128dw(6), pad_en, data_size=4B
  g1[1] = (int)(128u << 16);      // tensor_dim0 = 128 (bits 79:48 low half)
  g1[2] = (int)(8192u << 16);     // tensor_dim1 = 8192 (bits 111:80 low half)
  g1[3] = (int)(128u << 16);      // tile_dim0 = 128
  g1[4] = 16;                     // tile_dim1 = #gather indices
  g1[5] = 128;                    // tensor_dim0_stride = 128
  g1[6] = 0;
  g1[7] = 0;
  v4i g2, g3;
  g2[0] = (rows[0]  & 0xffff) | (rows[1]  << 16);
  g2[1] = (rows[2]  & 0xffff) | (rows[3]  << 16);
  g2[2] = (rows[4]  & 0xffff) | (rows[5]  << 16);
  g2[3] = (rows[6]  & 0xffff) | (rows[7]  << 16);
  g3[0] = (rows[8]  & 0xffff) | (rows[9]  << 16);
  g3[1] = (rows[10] & 0xffff) | (rows[11] << 16);
  g3[2] = (rows[12] & 0xffff) | (rows[13] << 16);
  g3[3] = (rows[14] & 0xffff) | (rows[15] << 16);
  v8i g4 = {};                    // extra operand on clang-23 lane; zero-filled
  __builtin_amdgcn_tensor_load_to_lds(g0, g1, g2, g3, g4, 0);
}

// Wave-uniform fetch of one tile's 16 indices; returns validity mask, clamps rows.
__device__ __forceinline__ unsigned load_tile_idx(const long long* __restrict__ topk,
                                                  int s, int tile, int* rows) {
  const long long* tp = topk + (size_t)s * kTopK + (size_t)tile * kKTile;
  unsigned vm = 0;
  #pragma unroll
  for (int k = 0; k < 16; ++k) {
    long long r = tp[k];
    rows[k] = (r >= 0) ? (int)r : 0;
    vm |= (unsigned)(r >= 0) << k;
  }
  return vm;
}

__launch_bounds__(kWaves * 32)
__global__ void sparse_topk_attn_f32wmma_tdm(
    const float* __restrict__ q,          // [S,H,D]
    const float* __restrict__ kv,         // [NKV,D]
    const long long* __restrict__ topk,   // [S,TOPK]
    float* __restrict__ out)              // [S,H,D]
{
  __shared__ __align__(16) float smem[kSmemF];  // tile buffers; merge aliases after barrier

  const int s    = blockIdx.x;
  const int lane = threadIdx.x & 31;
  const int wave = threadIdx.x >> 5;
  const int h    = lane & 15;     // head column (C-layout)
  const int half = lane >> 4;

  const float sm_scale = 0.08838834764831845f;  // 1/sqrt(128)
  const float NEG_INF  = -__builtin_huge_valf();

  // ---- Q^T as 32 B-matrix chunks, VGPR-resident ----
  v2f qb[kDChunk];
  {
    const float* qrow = q + ((size_t)s * kH + h) * kD + 2 * half;
    #pragma unroll
    for (int c = 0; c < kDChunk; ++c)
      qb[c] = *(const v2f*)(qrow + 4 * c);
  }

  const v8f vz = {};
  v8f oacc[kDTiles];
  #pragma unroll
  for (int t = 0; t < kDTiles; ++t) oacc[t] = vz;

  float m = NEG_INF, lsum = 0.f;

  const unsigned bufByte0 = (unsigned)(wave * 2 * kBufF) * 4u;
  constexpr int nIter = kTiles / kWaves;   // 8 tiles per wave

  // prologue: issue TDM gather for this wave's first tile
  unsigned maskCur;
  {
    int rows[16];
    maskCur = load_tile_idx(topk, s, wave, rows);
    asm volatile("" ::: "memory");
    tdm_gather16(bufByte0, kv, rows);
  }

  for (int it = 0; it < nIter; ++it) {
    const int buf = it & 1;

    // prefetch next tile: pack indices, issue TDM into the other buffer
    unsigned maskNext = 0;
    if (it + 1 < nIter) {
      int rn[16];
      maskNext = load_tile_idx(topk, s, wave + (it + 1) * kWaves, rn);
      asm volatile("s_wait_dscnt 0x0" ::: "memory");  // prior reads of target buf drained
      tdm_gather16(bufByte0 + (unsigned)((it + 1) & 1) * (unsigned)kBufF * 4u, kv, rn);
      __builtin_amdgcn_s_wait_tensorcnt(1);           // current tile's gather done
    } else {
      __builtin_amdgcn_s_wait_tensorcnt(0);
    }
    asm volatile("" ::: "memory");

    const float* kbuf = smem + (size_t)wave * 2 * kBufF + (size_t)buf * kBufF;

    // ---- S^T = K_tile x Q^T : 32 chained f32 WMMAs, A from LDS (conflict-free b64) ----
    v8f sacc = vz;
    {
      const float* kb = kbuf + h * kPitch + 2 * half;
      #pragma unroll
      for (int c = 0; c < kDChunk; ++c) {
        v2f a = *(const v2f*)(kb + 4 * c);
        sacc = __builtin_amdgcn_wmma_f32_16x16x4_f32(
            false, a, false, qb[c], (short)0, sacc, false, false);
      }
    }

    // ---- online softmax (column = head, per lane) ----
    float sreg[8];
    float tmax = NEG_INF;
    if (maskCur == 0xffffu) {                    // uniform fast path
      #pragma unroll
      for (int r = 0; r < 8; ++r) {
        float sv = sacc[r] * sm_scale;
        sreg[r] = sv;
        tmax = fmaxf(tmax, sv);
      }
    } else {
      unsigned mk = maskCur >> (half << 3);
      #pragma unroll
      for (int r = 0; r < 8; ++r) {
        float sv = ((mk >> r) & 1u) ? sacc[r] * sm_scale : NEG_INF;
        sreg[r] = sv;
        tmax = fmaxf(tmax, sv);
      }
    }
    tmax = fmaxf(tmax, __shfl_xor(tmax, 16, 32));

    float mnew  = fmaxf(m, tmax);
    float alpha = (mnew == NEG_INF) ? 1.f : __expf(m - mnew);
    float tsum  = 0.f;
    #pragma unroll
    for (int r = 0; r < 8; ++r) {
      float p = __expf(sreg[r] - mnew);
      sreg[r] = p;
      tsum += p;
    }
    tsum += __shfl_xor(tsum, 16, 32);
    lsum = lsum * alpha + tsum;
    m = mnew;

    #pragma unroll
    for (int t = 0; t < kDTiles; ++t)
      #pragma unroll
      for (int r = 0; r < 8; ++r)
        oacc[t][r] *= alpha;

    // ---- P^T C-layout -> 4 B chunks (8 shfl_xor) ----
    v2f pb[4];
    {
      float t0 = __shfl_xor(sreg[2], 16, 32);
      float t1 = __shfl_xor(sreg[3], 16, 32);
      float t2 = __shfl_xor(sreg[6], 16, 32);
      float t3 = __shfl_xor(sreg[7], 16, 32);
      float t4 = __shfl_xor(sreg[0], 16, 32);
      float t5 = __shfl_xor(sreg[1], 16, 32);
      float t6 = __shfl_xor(sreg[4], 16, 32);
      float t7 = __shfl_xor(sreg[5], 16, 32);
      pb[0].x = half ? t0 : sreg[0];  pb[0].y = half ? t1 : sreg[1];
      pb[1].x = half ? t2 : sreg[4];  pb[1].y = half ? t3 : sreg[5];
      pb[2].x = half ? sreg[2] : t4;  pb[2].y = half ? sreg[3] : t5;
      pb[3].x = half ? sreg[6] : t6;  pb[3].y = half ? sreg[7] : t7;
    }

    // ---- O^T += V^T x P^T : A-chunks via ds_read2-style pairs from LDS ----
    #pragma unroll
    for (int c = 0; c < 4; ++c) {
      const float* vc = kbuf + (4 * c + 2 * half) * kPitch + h;
      #pragma unroll
      for (int dt = 0; dt < kDTiles; ++dt) {
        v2f a;
        a.x = vc[dt * 16];            // V^T(d, key 4c+2*half)
        a.y = vc[kPitch + dt * 16];   // V^T(d, key 4c+1+2*half)
        oacc[dt] = __builtin_amdgcn_wmma_f32_16x16x4_f32(
            false, a, false, pb[c], (short)0, oacc[dt], false, false);
      }
    }

    maskCur = maskNext;
  }

  // ---- cross-wave merge: alias tile buffers as merge scratch (behind barrier) ----
  __syncthreads();
  float* mptr = smem;                 // [4][16]
  float* lptr = smem + 64;            // [4][16]
  float* optr = smem + 128;           // [4][8][8][32]

  if (lane < 16) { mptr[wave * 16 + lane] = m; lptr[wave * 16 + lane] = lsum; }
  #pragma unroll
  for (int dt = 0; dt < kDTiles; ++dt)
    #pragma unroll
    for (int r = 0; r < 8; ++r)
      optr[(((size_t)wave * kDTiles + dt) * 8 + r) * 32 + lane] = oacc[dt][r];
  __syncthreads();

  float mw[kWaves], mstar = NEG_INF;
  #pragma unroll
  for (int w = 0; w < kWaves; ++w) {
    mw[w] = mptr[w * 16 + h];
    mstar = fmaxf(mstar, mw[w]);
  }
  float sw[kWaves], lstar = 0.f;
  #pragma unroll
  for (int w = 0; w < kWaves; ++w) {
    float sc = (mw[w] == NEG_INF) ? 0.f : __expf(mw[w] - mstar);
    sw[w] = sc;
    lstar += sc * lptr[w * 16 + h];
  }
  const float inv = 1.f / lstar;

  #pragma unroll
  for (int i = 0; i < kDTiles / kWaves; ++i) {
    const int dt = wave * (kDTiles / kWaves) + i;
    #pragma unroll
    for (int r = 0; r < 8; ++r) {
      float v = 0.f;
      #pragma unroll
      for (int w = 0; w < kWaves; ++w)
        v += sw[w] * optr[(((size_t)w * kDTiles + dt) * 8 + r) * 32 + lane];
      const int d = dt * 16 + r + (half << 3);
      out[((size_t)s * kH + h) * kD + d] = v * inv;
    }
  }
}

extern "C" void kernel_launch(void* const* d_in, const int* in_sizes, int n_in,
                              void* d_out, int out_size, void* d_ws, size_t ws_size,
                              hipStream_t stream) {
  const float* q            = (const float*)d_in[0];       // [1,2048,16,128] f32
  const float* kv           = (const float*)d_in[1];       // [1,8192,128]    f32
  const long long* topk_idx = (const long long*)d_in[2];   // [1,2048,512]    i64
  float* out                = (float*)d_out;               // [1,2048,16,128] f32
  (void)in_sizes; (void)n_in; (void)out_size; (void)d_ws; (void)ws_size;

  dim3 grid(kS);
  dim3 block(kWaves * 32);
  hipLaunchKernelGGL(sparse_topk_attn_f32wmma_tdm, grid, block, 0, stream,
                     q, kv, topk_idx, out);
}